// CrossNetwork_15118284882274
// MI455X (gfx1250) — compile-verified
//
#include <hip/hip_runtime.h>

typedef __attribute__((ext_vector_type(2))) float v2f;
typedef __attribute__((ext_vector_type(4))) float v4f;
typedef __attribute__((ext_vector_type(8))) float v8f;

#define D_DIM 1024
#define ROWS_PER_BLOCK 128
#define THREADS 256
#define WROW 1028  // padded LDS row stride (floats): conflict-free b64 reads across lanes

// DCN cross network, 4 layers, collapsed to rank-1 form:
//   cur_i = a_i * x0 + sum_{j<i} b_j
//   s_i   = a_i * p_i + sum_{j<i} q[j][i],   p_i = x0 . w_i,  q[j][i] = b_j . w_i
//   a_{i+1} = a_i + s_i ;  out = a_4 * x0 + (b0+b1+b2+b3)
// p_i computed with V_WMMA_F32_16X16X4_F32 (16 rows per wave, K-loop over D).
__global__ __launch_bounds__(THREADS)
void dcn_cross_fused(const float* __restrict__ x,
                     const float* __restrict__ weights,
                     const float* __restrict__ biases,
                     float* __restrict__ out) {
    __shared__ float wpad[5 * WROW];            // rows 0-3: w_i ; row 4: zeros (B cols 4..15)
    __shared__ float bsv[D_DIM];                // b0+b1+b2+b3
    __shared__ float qm[16];                    // qm[j*4+i] = b_j . w_i
    __shared__ float ptile[ROWS_PER_BLOCK * 4]; // p[row][i]

    const int tid  = threadIdx.x;
    const int wave = tid >> 5;
    const int lane = tid & 31;

    // ---- stage weights into padded LDS (+ zeroed dummy row) ----
    for (int idx = tid; idx < 4 * D_DIM; idx += THREADS) {
        int r = idx >> 10, c = idx & (D_DIM - 1);
        wpad[r * WROW + c] = weights[idx];
    }
    for (int idx = tid; idx < WROW; idx += THREADS) wpad[4 * WROW + idx] = 0.0f;

    // ---- bias sum ----
    for (int d = tid; d < D_DIM; d += THREADS)
        bsv[d] = biases[d] + biases[D_DIM + d] + biases[2 * D_DIM + d] + biases[3 * D_DIM + d];

    // ---- q matrix: 16 length-1024 dots, 2 per wave, shuffle-reduced (wave32) ----
    for (int t = 0; t < 2; ++t) {
        int p = wave * 2 + t;
        int j = p >> 2, i = p & 3;
        float acc = 0.0f;
        for (int d = lane; d < D_DIM; d += 32)
            acc = fmaf(biases[j * D_DIM + d], weights[i * D_DIM + d], acc);
        for (int off = 16; off > 0; off >>= 1) acc += __shfl_xor(acc, off, 32);
        if (lane == 0) qm[p] = acc;
    }
    __syncthreads();

    // ---- WMMA phase: per wave, P[16x16] (cols 0-3 valid) = X[16xK] * W^T[Kx16] ----
    // A layout (16x4 f32): lane m   -> A[m][0..1] , lane m+16 -> A[m][2..3]
    // B layout ( 4x16 f32): lane n  -> B[0..1][n] , lane n+16 -> B[2..3][n]
    const int rowbase = blockIdx.x * ROWS_PER_BLOCK + wave * 16;
    const int r16 = lane & 15;
    const int hi  = lane >> 4;
    const float* gA = x + (size_t)(rowbase + r16) * D_DIM + 2 * hi;
    const float* lB = &wpad[(r16 < 4 ? r16 : 4) * WROW + 2 * hi]; // dummy row -> zeros, broadcast read

    v8f c = {};
    #pragma unroll 4
    for (int k = 0; k < D_DIM / 4; ++k) {
        v2f a = *(const v2f*)(gA + 4 * k);
        v2f b = *(const v2f*)(lB + 4 * k);
        c = __builtin_amdgcn_wmma_f32_16x16x4_f32(false, a, false, b, (short)0, c,
                                                  false, false);
    }
    // D layout: VGPR v -> (M=v, N=lane) lanes 0-15 ; (M=v+8, N=lane-16) lanes 16-31
    if (r16 < 4) {
        #pragma unroll
        for (int v = 0; v < 8; ++v)
            ptile[(wave * 16 + v + 8 * hi) * 4 + r16] = c[v];
    }
    __syncthreads();

    // ---- scalar recursion + axpy (second X read hits L2: same tile just streamed) ----
    const float q01 = qm[1];
    const float qs2 = qm[2] + qm[6];
    const float qs3 = qm[3] + qm[7] + qm[11];
    for (int m = 0; m < 16; ++m) {
        const float* pr = &ptile[(wave * 16 + m) * 4];
        float p0 = pr[0], p1 = pr[1], p2 = pr[2], p3 = pr[3];
        float a4 = 1.0f;
        float s = a4 * p0;          a4 += s;
        s = fmaf(a4, p1, q01);      a4 += s;
        s = fmaf(a4, p2, qs2);      a4 += s;
        s = fmaf(a4, p3, qs3);      a4 += s;
        const float* xr   = x   + (size_t)(rowbase + m) * D_DIM;
        float*       orow = out + (size_t)(rowbase + m) * D_DIM;
        #pragma unroll
        for (int cc = 0; cc < 8; ++cc) {
            int idx = cc * 128 + lane * 4;
            v4f xv = *(const v4f*)(xr + idx);
            v4f bs = *(const v4f*)(&bsv[idx]);
            v4f o;
            o.x = fmaf(a4, xv.x, bs.x);
            o.y = fmaf(a4, xv.y, bs.y);
            o.z = fmaf(a4, xv.z, bs.z);
            o.w = fmaf(a4, xv.w, bs.w);
            __builtin_nontemporal_store(o, (v4f*)(orow + idx)); // don't evict X from L2
        }
    }
}

extern "C" void kernel_launch(void* const* d_in, const int* in_sizes, int n_in,
                              void* d_out, int out_size, void* d_ws, size_t ws_size,
                              hipStream_t stream) {
    const float* x = (const float*)d_in[0];
    const float* w = (const float*)d_in[1];
    const float* b = (const float*)d_in[2];
    float* out = (float*)d_out;
    const int B = in_sizes[0] / D_DIM;           // 32768
    dim3 grid(B / ROWS_PER_BLOCK);               // 256 blocks of 256 threads (8 waves)
    hipLaunchKernelGGL(dcn_cross_fused, grid, dim3(THREADS), 0, stream, x, w, b, out);
}